// KPlexPool_39273180955454
// MI455X (gfx1250) — compile-verified
//
#include <hip/hip_runtime.h>
#include <hip/hip_bf16.h>
#include <stdint.h>

// Problem constants (match reference)
#define NN   100000
#define EE   1600000
#define FIN  128
#define HH   128
#define MMC  150000
#define KCC  60000
#define BBG  64
#define COUT 10
#define BN_EPS 1e-5f

typedef __attribute__((ext_vector_type(2))) float v2f;
typedef __attribute__((ext_vector_type(8))) float v8f;

// ---------------- atomics ----------------
__device__ __forceinline__ void atomAddF(float* p, float v) {
    __hip_atomic_fetch_add(p, v, __ATOMIC_RELAXED, __HIP_MEMORY_SCOPE_AGENT);
}
// valid for non-negative floats with non-negative init (post-ReLU values)
__device__ __forceinline__ void atomMaxNonneg(float* p, float v) {
    __hip_atomic_fetch_max((int*)p, __float_as_int(v), __ATOMIC_RELAXED, __HIP_MEMORY_SCOPE_AGENT);
}

// ---------------- generic ----------------
__global__ void k_zero(float* __restrict__ p, long long n) {
    long long i = (long long)blockIdx.x * blockDim.x + threadIdx.x;
    long long stride = (long long)gridDim.x * blockDim.x;
    for (; i < n; i += stride) p[i] = 0.0f;
}

// dst = rsqrt(src + 1) in place (deg -> dinv)
__global__ void k_rsqrt1p(float* __restrict__ p, int n) {
    int i = blockIdx.x * blockDim.x + threadIdx.x;
    if (i < n) p[i] = rsqrtf(p[i] + 1.0f);
}

// ---------------- WMMA f32 GEMM: C[Mr,128] = A[Mr,K] @ B[K,128] ----------------
// grid.x = Mr/16 blocks; 256 threads = 8 waves; wave w owns output cols [16w,16w+16)
__global__ void k_wmma_gemm(const float* __restrict__ A, const float* __restrict__ Bm,
                            float* __restrict__ C, int K) {
    const int lane = threadIdx.x & 31;
    const int wave = threadIdx.x >> 5;
    const int half = lane >> 4;      // 0: lanes 0-15, 1: lanes 16-31
    const int lr   = lane & 15;
    const long long r0 = (long long)blockIdx.x * 16;
    const int c0 = wave * 16;

    v8f acc = {};
    for (int k0 = 0; k0 < K; k0 += 4) {
        // A 16x4 fragment: VGPR0 = {K0 | K2}, VGPR1 = {K1 | K3} (ISA 7.12.2)
        v2f a, b;
        const float* arow = A + (r0 + lr) * (long long)K + (k0 + 2 * half);
        a.x = arow[0];
        a.y = arow[1];
        // B 4x16 fragment, mirrored layout on N
        const float* bk = Bm + (long long)(k0 + 2 * half) * 128 + c0 + lr;
        b.x = bk[0];
        b.y = bk[128];
        acc = __builtin_amdgcn_wmma_f32_16x16x4_f32(false, a, false, b,
                                                    (short)0, acc, false, false);
    }
    // C/D layout: VGPR v -> row v (lanes 0-15) / row v+8 (lanes 16-31), col = lr
#pragma unroll
    for (int v = 0; v < 8; ++v) {
        C[(r0 + v + 8 * half) * 128 + c0 + lr] = acc[v];
    }
}

// ---------------- scalar scatters ----------------
__global__ void k_deg_scatter(const int* __restrict__ dst, const float* __restrict__ w,
                              float* __restrict__ deg, int E) {
    int e = blockIdx.x * blockDim.x + threadIdx.x;
    if (e < E) atomAddF(&deg[dst[e]], w[e]);
}
__global__ void k_count_scatter(const int* __restrict__ idx, float* __restrict__ out, int n) {
    int i = blockIdx.x * blockDim.x + threadIdx.x;
    if (i < n) atomAddF(&out[idx[i]], 1.0f);
}
// out[oidx[i]] += in[iidx[i]]
__global__ void k_gs_scalar(const int* __restrict__ oidx, const int* __restrict__ iidx,
                            const float* __restrict__ in, float* __restrict__ out, int n) {
    int i = blockIdx.x * blockDim.x + threadIdx.x;
    if (i < n) atomAddF(&out[oidx[i]], in[iidx[i]]);
}
// ye1[dst] += w * zn1[src]
__global__ void k_edge_scalar(const int* __restrict__ src, const int* __restrict__ dst,
                              const float* __restrict__ w, const float* __restrict__ zn1,
                              float* __restrict__ ye1, int E) {
    int e = blockIdx.x * blockDim.x + threadIdx.x;
    if (e < E) atomAddF(&ye1[dst[e]], w[e] * zn1[src[e]]);
}

// ---------------- row (128-feature) kernels ----------------
// agg init: Y[i,:] = dinv[i]^2 * X[i,:]
__global__ void k_selfloop_init(const float* __restrict__ X, const float* __restrict__ dinv,
                                float* __restrict__ Y, long long total) {
    long long idx = (long long)blockIdx.x * blockDim.x + threadIdx.x;
    if (idx >= total) return;
    long long i = idx >> 7;
    float d = dinv[i];
    Y[idx] = d * d * X[idx];
}

// Y[dst,:] += coef * X[src,:] ; coef = w (* dinv[src]*dinv[dst] if dinv != null)
// one 32-thread group per edge, float4 per thread
__global__ void k_edge_scatter(const int* __restrict__ src, const int* __restrict__ dst,
                               const float* __restrict__ w, const float* __restrict__ dinv,
                               const float* __restrict__ X, float* __restrict__ Y, int E) {
    long long t = (long long)blockIdx.x * blockDim.x + threadIdx.x;
    int e = (int)(t >> 5);
    if (e >= E) return;
    int lane = (int)(t & 31);
    int s = src[e], d = dst[e];
    float coef = w[e];
    if (dinv) coef *= dinv[s] * dinv[d];
    const float4 v = *reinterpret_cast<const float4*>(X + (long long)s * 128 + lane * 4);
    float* o = Y + (long long)d * 128 + lane * 4;
    atomAddF(o + 0, coef * v.x);
    atomAddF(o + 1, coef * v.y);
    atomAddF(o + 2, coef * v.z);
    atomAddF(o + 3, coef * v.w);
}

__global__ void k_bias_relu(float* __restrict__ h, const float* __restrict__ b, long long total) {
    long long idx = (long long)blockIdx.x * blockDim.x + threadIdx.x;
    if (idx >= total) return;
    int f = (int)(idx & 127);
    h[idx] = fmaxf(h[idx] + b[f], 0.0f);
}

// node pooling: zbuf[g, 0:128] += h[i,:] (sum), zbuf[g, 128:256] max= h[i,:]
__global__ void k_node_pool(const float* __restrict__ h, const int* __restrict__ batch,
                            float* __restrict__ zbuf, int n) {
    long long t = (long long)blockIdx.x * blockDim.x + threadIdx.x;
    int i = (int)(t >> 5);
    if (i >= n) return;
    int lane = (int)(t & 31);
    int g = batch[i];
    const float4 v = *reinterpret_cast<const float4*>(h + (long long)i * 128 + lane * 4);
    float* ps = zbuf + (long long)g * 512 + lane * 4;
    atomAddF(ps + 0, v.x); atomAddF(ps + 1, v.y);
    atomAddF(ps + 2, v.z); atomAddF(ps + 3, v.w);
    float* pm = zbuf + (long long)g * 512 + 128 + lane * 4;
    atomMaxNonneg(pm + 0, v.x); atomMaxNonneg(pm + 1, v.y);
    atomMaxNonneg(pm + 2, v.z); atomMaxNonneg(pm + 3, v.w);
}

// cover pooling into xp[KC,256]: cols 0:128 add, 128:256 max
__global__ void k_cover_pool(const float* __restrict__ h, const int* __restrict__ nodes,
                             const int* __restrict__ clusters, float* __restrict__ xp, int m) {
    long long t = (long long)blockIdx.x * blockDim.x + threadIdx.x;
    int j = (int)(t >> 5);
    if (j >= m) return;
    int lane = (int)(t & 31);
    int n = nodes[j], c = clusters[j];
    const float4 v = *reinterpret_cast<const float4*>(h + (long long)n * 128 + lane * 4);
    float* pa = xp + (long long)c * 256 + lane * 4;
    atomAddF(pa + 0, v.x); atomAddF(pa + 1, v.y);
    atomAddF(pa + 2, v.z); atomAddF(pa + 3, v.w);
    float* pm = xp + (long long)c * 256 + 128 + lane * 4;
    atomMaxNonneg(pm + 0, v.x); atomMaxNonneg(pm + 1, v.y);
    atomMaxNonneg(pm + 2, v.z); atomMaxNonneg(pm + 3, v.w);
}

// out[oidx[j],:] += in[iidx[j],:]
__global__ void k_scatter_rows(const int* __restrict__ oidx, const int* __restrict__ iidx,
                               const float* __restrict__ in, float* __restrict__ out, int m) {
    long long t = (long long)blockIdx.x * blockDim.x + threadIdx.x;
    int j = (int)(t >> 5);
    if (j >= m) return;
    int lane = (int)(t & 31);
    const float4 v = *reinterpret_cast<const float4*>(in + (long long)iidx[j] * 128 + lane * 4);
    float* o = out + (long long)oidx[j] * 128 + lane * 4;
    atomAddF(o + 0, v.x); atomAddF(o + 1, v.y);
    atomAddF(o + 2, v.z); atomAddF(o + 3, v.w);
}

// X[i,:] *= dinv[i]
__global__ void k_scale_rows(float* __restrict__ X, const float* __restrict__ dinv, long long total) {
    long long idx = (long long)blockIdx.x * blockDim.x + threadIdx.x;
    if (idx >= total) return;
    X[idx] *= dinv[idx >> 7];
}

// hp = relu(dinvp*(ap + xws) + b)  (in place on ap)
__global__ void k_hp(float* __restrict__ ap, const float* __restrict__ xws,
                     const float* __restrict__ dinvp, const float* __restrict__ b, long long total) {
    long long idx = (long long)blockIdx.x * blockDim.x + threadIdx.x;
    if (idx >= total) return;
    long long c = idx >> 7;
    int f = (int)(idx & 127);
    ap[idx] = fmaxf(dinvp[c] * (ap[idx] + xws[idx]) + b[f], 0.0f);
}

// cluster pooling: zbuf[g,256:384] += hp, zbuf[g,384:512] max= hp
__global__ void k_cluster_pool(const float* __restrict__ hp, const int* __restrict__ cb,
                               float* __restrict__ zbuf, int n) {
    long long t = (long long)blockIdx.x * blockDim.x + threadIdx.x;
    int c = (int)(t >> 5);
    if (c >= n) return;
    int lane = (int)(t & 31);
    int g = cb[c];
    const float4 v = *reinterpret_cast<const float4*>(hp + (long long)c * 128 + lane * 4);
    float* ps = zbuf + (long long)g * 512 + 256 + lane * 4;
    atomAddF(ps + 0, v.x); atomAddF(ps + 1, v.y);
    atomAddF(ps + 2, v.z); atomAddF(ps + 3, v.w);
    float* pm = zbuf + (long long)g * 512 + 384 + lane * 4;
    atomMaxNonneg(pm + 0, v.x); atomMaxNonneg(pm + 1, v.y);
    atomMaxNonneg(pm + 2, v.z); atomMaxNonneg(pm + 3, v.w);
}

// ---------------- head: BN -> W1+relu -> W2 -> softmax. 1 block per graph ----------------
__global__ void k_head(const float* __restrict__ zbuf, const float* __restrict__ cc,
                       const float* __restrict__ g, const float* __restrict__ bta,
                       const float* __restrict__ mu, const float* __restrict__ var,
                       const float* __restrict__ W1, const float* __restrict__ b1,
                       const float* __restrict__ W2, const float* __restrict__ b2,
                       float* __restrict__ out) {
    __shared__ float zrow[512];
    __shared__ float z1[128];
    __shared__ float logits[COUT];
    const int b = blockIdx.x;
    const int t = threadIdx.x;
    const float inv_cnt = 1.0f / cc[b];
    for (int j = t; j < 512; j += 128) {
        float v = zbuf[(long long)b * 512 + j];
        if (j >= 256 && j < 384) v *= inv_cnt;  // mean pool
        v = (v - mu[j]) * rsqrtf(var[j] + BN_EPS) * g[j] + bta[j];
        zrow[j] = v;
    }
    __syncthreads();
    float acc = b1[t];
    for (int j = 0; j < 512; ++j) acc = fmaf(zrow[j], W1[j * 128 + t], acc);
    z1[t] = fmaxf(acc, 0.0f);
    __syncthreads();
    if (t < COUT) {
        float a2 = b2[t];
        for (int j = 0; j < 128; ++j) a2 = fmaf(z1[j], W2[j * COUT + t], a2);
        logits[t] = a2;
    }
    __syncthreads();
    if (t == 0) {
        float mx = logits[0];
        for (int k = 1; k < COUT; ++k) mx = fmaxf(mx, logits[k]);
        float s = 0.0f, ex[COUT];
        for (int k = 0; k < COUT; ++k) { ex[k] = __expf(logits[k] - mx); s += ex[k]; }
        float inv = 1.0f / s;
        for (int k = 0; k < COUT; ++k) out[b * COUT + k] = ex[k] * inv;
    }
}

// ---------------- host side ----------------
#define L1D(kern, n, ...) do {                                                    \
    long long _n = (long long)(n);                                                \
    long long _g = (_n + 255) / 256;                                              \
    kern<<<dim3((unsigned)_g), dim3(256), 0, stream>>>(__VA_ARGS__);              \
} while (0)

extern "C" void kernel_launch(void* const* d_in, const int* in_sizes, int n_in,
                              void* d_out, int out_size, void* d_ws, size_t ws_size,
                              hipStream_t stream) {
    (void)in_sizes; (void)n_in; (void)out_size; (void)ws_size;

    const float* x        = (const float*)d_in[0];
    const int*   eidx     = (const int*)d_in[1];
    const int*   src      = eidx;
    const int*   dst      = eidx + EE;
    const float* w        = (const float*)d_in[2];
    const int*   batch    = (const int*)d_in[3];
    const int*   cnodes   = (const int*)d_in[4];
    const int*   cclust   = (const int*)d_in[5];
    const int*   cbatch   = (const int*)d_in[6];
    const float* W_in     = (const float*)d_in[7];
    const float* b_in     = (const float*)d_in[8];
    const float* W_blk    = (const float*)d_in[9];
    const float* b_blk    = (const float*)d_in[10];
    const float* bn_g     = (const float*)d_in[11];
    const float* bn_b     = (const float*)d_in[12];
    const float* bn_m     = (const float*)d_in[13];
    const float* bn_v     = (const float*)d_in[14];
    const float* W1       = (const float*)d_in[15];
    const float* b1       = (const float*)d_in[16];
    const float* W2       = (const float*)d_in[17];
    const float* b2       = (const float*)d_in[18];
    float* out = (float*)d_out;

    // workspace carve-up (~227 MB)
    float* xw   = (float*)d_ws;                 // N*128  (later reused as zn)
    float* h    = xw   + (size_t)NN * 128;      // N*128  (agg -> h, later reused as ye)
    float* deg  = h    + (size_t)NN * 128;      // N      (-> dinv)
    float* xp   = deg  + NN;                    // KC*256
    float* xws  = xp   + (size_t)KCC * 256;     // KC*128
    float* ap   = xws  + (size_t)KCC * 128;     // KC*128 (-> hp)
    float* degp = ap   + (size_t)KCC * 128;     // KC     (-> dinvp)
    float* zn1  = degp + KCC;                   // N
    float* ye1  = zn1  + NN;                    // N
    float* zbuf = ye1  + NN;                    // B*512
    float* cc   = zbuf + (size_t)BBG * 512;     // B

    // 1) weighted degrees + dinv
    L1D(k_zero, NN, deg, (long long)NN);
    L1D(k_deg_scatter, EE, dst, w, deg, EE);
    L1D(k_rsqrt1p, NN, deg, NN);                // deg now holds dinv

    // 2) xw = x @ W_in   (WMMA f32)
    k_wmma_gemm<<<dim3(NN / 16), dim3(256), 0, stream>>>(x, W_in, xw, FIN);

    // 3) agg = dinv^2*xw ; += edge messages ; h = relu(agg + b_in)
    L1D(k_selfloop_init, (long long)NN * 128, xw, deg, h, (long long)NN * 128);
    L1D(k_edge_scatter, (long long)EE * 32, src, dst, w, deg, xw, h, EE);
    L1D(k_bias_relu, (long long)NN * 128, h, b_in, (long long)NN * 128);

    // 4) node pooling (sum cols 0:128, max cols 128:256 of zbuf)
    L1D(k_zero, (long long)BBG * 512, zbuf, (long long)BBG * 512);
    L1D(k_zero, BBG, cc, (long long)BBG);
    L1D(k_node_pool, (long long)NN * 32, h, batch, zbuf, NN);

    // 5) cover pool -> xp[KC,256] (add | max)
    L1D(k_zero, (long long)KCC * 256, xp, (long long)KCC * 256);
    L1D(k_cover_pool, (long long)MMC * 32, h, cnodes, cclust, xp, MMC);

    // 6) degp = aprime(ones) + 1 ; dinvp = rsqrt
    L1D(k_zero, NN, zn1, (long long)NN);
    L1D(k_zero, NN, ye1, (long long)NN);
    L1D(k_zero, KCC, degp, (long long)KCC);
    L1D(k_count_scatter, MMC, cnodes, zn1, MMC);
    L1D(k_edge_scalar, EE, src, dst, w, zn1, ye1, EE);
    L1D(k_gs_scalar, MMC, cclust, cnodes, ye1, degp, MMC);
    L1D(k_rsqrt1p, KCC, degp, KCC);             // degp now holds dinvp

    // 7) xws = dinvp * (xp @ W_blk)   (WMMA f32, K=256)
    k_wmma_gemm<<<dim3(KCC / 16), dim3(256), 0, stream>>>(xp, W_blk, xws, 2 * HH);
    L1D(k_scale_rows, (long long)KCC * 128, xws, degp, (long long)KCC * 128);

    // 8) aprime(xws):  zn = C xws ; ye = A zn ; ap = C^T ye
    L1D(k_zero, (long long)NN * 128, xw, (long long)NN * 128);       // zn := xw buffer
    L1D(k_scatter_rows, (long long)MMC * 32, cnodes, cclust, xws, xw, MMC);
    L1D(k_zero, (long long)NN * 128, h, (long long)NN * 128);        // ye := h buffer
    L1D(k_edge_scatter, (long long)EE * 32, src, dst, w, (const float*)nullptr, xw, h, EE);
    L1D(k_zero, (long long)KCC * 128, ap, (long long)KCC * 128);
    L1D(k_scatter_rows, (long long)MMC * 32, cclust, cnodes, h, ap, MMC);

    // 9) hp = relu(dinvp*(ap + xws) + b_blk)   (in place on ap)
    L1D(k_hp, (long long)KCC * 128, ap, xws, degp, b_blk, (long long)KCC * 128);

    // 10) cluster pooling (mean cols 256:384 via cc, max cols 384:512)
    L1D(k_count_scatter, KCC, cbatch, cc, KCC);
    L1D(k_cluster_pool, (long long)KCC * 32, ap, cbatch, zbuf, KCC);

    // 11) head: BN -> MLP -> softmax
    k_head<<<dim3(BBG), dim3(128), 0, stream>>>(zbuf, cc, bn_g, bn_b, bn_m, bn_v,
                                                W1, b1, W2, b2, out);
}